// GCNGAT_17446157156488
// MI455X (gfx1250) — compile-verified
//
#include <hip/hip_runtime.h>
#include <hip/hip_bf16.h>

typedef __attribute__((ext_vector_type(2))) float v2f;
typedef __attribute__((ext_vector_type(8))) float v8f;

#define N_NODES 50000
#define N_EDGES 1600000
#define F_IN    2000
#define HID     16
#define HEADS   4
#define CH      16
#define HC      64   // HEADS*CH
#define NCLS    3

// ---------------------------------------------------------------------------
__global__ void fill_kernel(float* __restrict__ p, long n, float v) {
    long i = (long)blockIdx.x * blockDim.x + threadIdx.x;
    if (i < n) p[i] = v;
}

// dst[c*K + k] = src[k*C + c]   (row-major K x C  ->  row-major C x K)
__global__ void transpose_kernel(const float* __restrict__ src, float* __restrict__ dst,
                                 int K, int C) {
    int i = blockIdx.x * blockDim.x + threadIdx.x;
    if (i >= K * C) return;
    int k = i / C, c = i - k * C;
    dst[c * K + k] = src[i];
}

// ---------------------------------------------------------------------------
// h0 = x @ W1   [50000,2000] @ [2000,16]  -- one wave per 16-row tile,
// V_WMMA_F32_16X16X4_F32 along K (500 steps). HBM-bound on x (400 MB).
// B comes from pre-transposed W1T[16][2000] so each step is one b64 load.
__global__ void gemm1_wmma(const float* __restrict__ x,
                           const float* __restrict__ W1T,
                           float* __restrict__ h0) {
    const int wave = threadIdx.x >> 5;
    const int lane = threadIdx.x & 31;
    const int tile = blockIdx.x * (blockDim.x >> 5) + wave;
    if (tile >= N_NODES / 16) return;            // wave-uniform guard (EXEC stays full)
    const int half = lane >> 4;                  // 0: K pair {0,1}, 1: K pair {2,3}
    const int r    = lane & 15;                  // A: row within tile; B: column
    const long rowOff = (long)(tile * 16 + r) * F_IN;
    const float* bcol = W1T + (long)r * F_IN;    // column r of W1, contiguous in K
    v8f acc = {};
    for (int k = 0; k < F_IN; k += 4) {
        const int kk = k + half * 2;
        v2f a = *(const v2f*)(x + rowOff + kk);                 // A[r, kk..kk+1]
        v2f b = *(const v2f*)(bcol + kk);                       // B[kk..kk+1, r]
        acc = __builtin_amdgcn_wmma_f32_16x16x4_f32(
                false, a, false, b, (short)0, acc, false, false);
    }
    const int rowBase = tile * 16;
#pragma unroll
    for (int j = 0; j < 8; ++j) {
        int m = j + half * 8;                                   // C/D layout
        h0[(long)(rowBase + m) * HID + r] = acc[j];
    }
}

// ---------------------------------------------------------------------------
// g = h1 @ Wg   [50000,16] @ [16,64]  -- wave per (rowTile, colTile), K=16.
// B from pre-transposed WgT[64][16].
__global__ void gemm2_wmma(const float* __restrict__ h1,
                           const float* __restrict__ WgT,
                           float* __restrict__ g) {
    const int wave = threadIdx.x >> 5;
    const int lane = threadIdx.x & 31;
    const int tile = blockIdx.x * (blockDim.x >> 5) + wave;
    if (tile >= (N_NODES / 16) * 4) return;      // wave-uniform guard
    const int rowTile = tile >> 2;
    const int colTile = tile & 3;
    const int half = lane >> 4;
    const int r    = lane & 15;
    const int rowBase = rowTile * 16;
    const float* bcol = WgT + (long)(colTile * 16 + r) * HID;   // column of Wg
    v8f acc = {};
#pragma unroll
    for (int k = 0; k < HID; k += 4) {
        const int kk = k + half * 2;
        v2f a = *(const v2f*)(h1 + (long)(rowBase + r) * HID + kk);
        v2f b = *(const v2f*)(bcol + kk);
        acc = __builtin_amdgcn_wmma_f32_16x16x4_f32(
                false, a, false, b, (short)0, acc, false, false);
    }
#pragma unroll
    for (int j = 0; j < 8; ++j) {
        int m = j + half * 8;
        g[(long)(rowBase + m) * HC + colTile * 16 + r] = acc[j];
    }
}

// ---------------------------------------------------------------------------
__global__ void deg_kernel(const int* __restrict__ ei, float* __restrict__ deg) {
    int e = blockIdx.x * blockDim.x + threadIdx.x;
    if (e < N_EDGES) atomicAdd(&deg[ei[N_EDGES + e]], 1.0f);
}

__global__ void dinv_kernel(float* __restrict__ d) {
    int n = blockIdx.x * blockDim.x + threadIdx.x;
    if (n < N_NODES) d[n] = rsqrtf(d[n] + 1.0f);   // +1 for the self loop
}

// hacc[dst] += h0[src] * dinv[src]*dinv[dst]   (edges only; self loop in finalize)
__global__ void gcn_agg(const int* __restrict__ ei, const float* __restrict__ dinv,
                        const float* __restrict__ h0, float* __restrict__ hacc) {
    int e = blockIdx.x * blockDim.x + threadIdx.x;
    if (e >= N_EDGES) return;
    const int s = ei[e], d = ei[N_EDGES + e];
    const float nrm = dinv[s] * dinv[d];
    const float4* hs = (const float4*)(h0 + (long)s * HID);
    float* od = hacc + (long)d * HID;
#pragma unroll
    for (int q = 0; q < 4; ++q) {
        float4 v = hs[q];
        atomicAdd(od + q * 4 + 0, v.x * nrm);
        atomicAdd(od + q * 4 + 1, v.y * nrm);
        atomicAdd(od + q * 4 + 2, v.z * nrm);
        atomicAdd(od + q * 4 + 3, v.w * nrm);
    }
}

// h1 = relu(hacc + self_loop + b1)
__global__ void gcn_fin(const float* __restrict__ h0, const float* __restrict__ dinv,
                        const float* __restrict__ b1, float* __restrict__ hacc) {
    int i = blockIdx.x * blockDim.x + threadIdx.x;
    if (i >= N_NODES * HID) return;
    const int n = i >> 4, c = i & 15;
    const float di = dinv[n];
    float v = hacc[i] + h0[i] * di * di + b1[c];
    hacc[i] = v > 0.f ? v : 0.f;
}

// ---------------------------------------------------------------------------
__global__ void att_coeff(const float* __restrict__ g, const float* __restrict__ asrc,
                          const float* __restrict__ adst,
                          float* __restrict__ a_s, float* __restrict__ a_d) {
    int i = blockIdx.x * blockDim.x + threadIdx.x;
    if (i >= N_NODES * HEADS) return;
    const int n = i >> 2, h = i & 3;
    const float* gr = g + (long)n * HC + h * CH;
    float s = 0.f, d = 0.f;
#pragma unroll
    for (int c = 0; c < CH; ++c) {
        s += gr[c] * asrc[h * CH + c];
        d += gr[c] * adst[h * CH + c];
    }
    a_s[i] = s; a_d[i] = d;
}

__device__ __forceinline__ float leaky(float v) { return v > 0.f ? v : 0.2f * v; }

__global__ void att_max(const int* __restrict__ ei, const float* __restrict__ a_s,
                        const float* __restrict__ a_d, float* __restrict__ m) {
    int e = blockIdx.x * blockDim.x + threadIdx.x;
    if (e >= N_EDGES + N_NODES) return;
    int s, d;
    if (e < N_EDGES) { s = ei[e]; d = ei[N_EDGES + e]; } else { s = d = e - N_EDGES; }
#pragma unroll
    for (int h = 0; h < HEADS; ++h)
        atomicMax(&m[d * HEADS + h], leaky(a_s[s * HEADS + h] + a_d[d * HEADS + h]));
}

__global__ void att_den(const int* __restrict__ ei, const float* __restrict__ a_s,
                        const float* __restrict__ a_d, const float* __restrict__ m,
                        float* __restrict__ den) {
    int e = blockIdx.x * blockDim.x + threadIdx.x;
    if (e >= N_EDGES + N_NODES) return;
    int s, d;
    if (e < N_EDGES) { s = ei[e]; d = ei[N_EDGES + e]; } else { s = d = e - N_EDGES; }
#pragma unroll
    for (int h = 0; h < HEADS; ++h) {
        float v = leaky(a_s[s * HEADS + h] + a_d[d * HEADS + h]);
        atomicAdd(&den[d * HEADS + h], __expf(v - m[d * HEADS + h]));
    }
}

// one thread per (edge, head); 16 channel atomics each
__global__ void att_agg(const int* __restrict__ ei, const float* __restrict__ a_s,
                        const float* __restrict__ a_d, const float* __restrict__ m,
                        const float* __restrict__ den, const float* __restrict__ g,
                        float* __restrict__ oacc) {
    long i = (long)blockIdx.x * blockDim.x + threadIdx.x;
    if (i >= (long)(N_EDGES + N_NODES) * HEADS) return;
    const int e = (int)(i >> 2), h = (int)(i & 3);
    int s, d;
    if (e < N_EDGES) { s = ei[e]; d = ei[N_EDGES + e]; } else { s = d = e - N_EDGES; }
    float v = leaky(a_s[s * HEADS + h] + a_d[d * HEADS + h]);
    float alpha = __expf(v - m[d * HEADS + h]) / (den[d * HEADS + h] + 1e-16f);
    const float* gs = g + (long)s * HC + h * CH;
    float* od = oacc + (long)d * HC + h * CH;
#pragma unroll
    for (int c = 0; c < CH; ++c) atomicAdd(od + c, gs[c] * alpha);
}

// ---------------------------------------------------------------------------
__global__ void head_kernel(const float* __restrict__ oacc, const float* __restrict__ bg,
                            const float* __restrict__ Wo, const float* __restrict__ bo,
                            float* __restrict__ out) {
    int n = blockIdx.x * blockDim.x + threadIdx.x;
    if (n >= N_NODES) return;
    float row[HC];
#pragma unroll
    for (int c = 0; c < HC; ++c) row[c] = oacc[(long)n * HC + c] + bg[c];
    float l[NCLS];
#pragma unroll
    for (int j = 0; j < NCLS; ++j) {
        float s = bo[j];
#pragma unroll
        for (int c = 0; c < HC; ++c) s += row[c] * Wo[c * NCLS + j];
        l[j] = s;
    }
    float mx = fmaxf(l[0], fmaxf(l[1], l[2]));
    float lse = logf(__expf(l[0] - mx) + __expf(l[1] - mx) + __expf(l[2] - mx));
#pragma unroll
    for (int j = 0; j < NCLS; ++j) out[(long)n * NCLS + j] = l[j] - mx - lse;
}

// ---------------------------------------------------------------------------
extern "C" void kernel_launch(void* const* d_in, const int* in_sizes, int n_in,
                              void* d_out, int out_size, void* d_ws, size_t ws_size,
                              hipStream_t stream) {
    const float* x    = (const float*)d_in[0];
    const int*   ei   = (const int*)  d_in[1];   // [2,E]: row0=src, row1=dst
    const float* W1   = (const float*)d_in[2];
    const float* b1   = (const float*)d_in[3];
    const float* Wg   = (const float*)d_in[4];
    const float* asrc = (const float*)d_in[5];
    const float* adst = (const float*)d_in[6];
    const float* bg   = (const float*)d_in[7];
    const float* Wo   = (const float*)d_in[8];
    const float* bo   = (const float*)d_in[9];
    float* out = (float*)d_out;

    float* ws   = (float*)d_ws;
    float* h0   = ws; ws += (long)N_NODES * HID;
    float* dinv = ws; ws += N_NODES;
    float* h1   = ws; ws += (long)N_NODES * HID;    // GCN accumulator -> relu'd h
    float* g    = ws; ws += (long)N_NODES * HC;
    float* a_s  = ws; ws += (long)N_NODES * HEADS;
    float* a_d  = ws; ws += (long)N_NODES * HEADS;
    float* mbuf = ws; ws += (long)N_NODES * HEADS;
    float* den  = ws; ws += (long)N_NODES * HEADS;
    float* oacc = ws; ws += (long)N_NODES * HC;
    float* W1T  = ws; ws += (long)F_IN * HID;       // [16][2000]
    float* WgT  = ws; ws += (long)HID * HC;         // [64][16]

    const int T = 256;
    auto nb = [](long n, int t) { return (int)((n + t - 1) / t); };

    // deterministic re-init of every accumulator
    fill_kernel<<<nb((long)N_NODES, T), T, 0, stream>>>(dinv, N_NODES, 0.f);
    fill_kernel<<<nb((long)N_NODES * HID, T), T, 0, stream>>>(h1, (long)N_NODES * HID, 0.f);
    fill_kernel<<<nb((long)N_NODES * HEADS, T), T, 0, stream>>>(mbuf, (long)N_NODES * HEADS, -3.0e38f);
    fill_kernel<<<nb((long)N_NODES * HEADS, T), T, 0, stream>>>(den, (long)N_NODES * HEADS, 0.f);
    fill_kernel<<<nb((long)N_NODES * HC, T), T, 0, stream>>>(oacc, (long)N_NODES * HC, 0.f);

    // weight transposes (tiny, L2-resident afterwards)
    transpose_kernel<<<nb((long)F_IN * HID, T), T, 0, stream>>>(W1, W1T, F_IN, HID);
    transpose_kernel<<<nb((long)HID * HC, T), T, 0, stream>>>(Wg, WgT, HID, HC);

    gemm1_wmma<<<nb(N_NODES / 16, 8), 256, 0, stream>>>(x, W1T, h0);
    deg_kernel<<<nb(N_EDGES, T), T, 0, stream>>>(ei, dinv);
    dinv_kernel<<<nb(N_NODES, T), T, 0, stream>>>(dinv);
    gcn_agg<<<nb(N_EDGES, T), T, 0, stream>>>(ei, dinv, h0, h1);
    gcn_fin<<<nb((long)N_NODES * HID, T), T, 0, stream>>>(h0, dinv, b1, h1);
    gemm2_wmma<<<nb((N_NODES / 16) * 4, 8), 256, 0, stream>>>(h1, WgT, g);
    att_coeff<<<nb((long)N_NODES * HEADS, T), T, 0, stream>>>(g, asrc, adst, a_s, a_d);
    att_max<<<nb(N_EDGES + N_NODES, T), T, 0, stream>>>(ei, a_s, a_d, mbuf);
    att_den<<<nb(N_EDGES + N_NODES, T), T, 0, stream>>>(ei, a_s, a_d, mbuf, den);
    att_agg<<<nb((long)(N_EDGES + N_NODES) * HEADS, T), T, 0, stream>>>(ei, a_s, a_d, mbuf, den, g, oacc);
    head_kernel<<<nb(N_NODES, T), T, 0, stream>>>(oacc, bg, Wo, bo, out);
}